// SS2D_8289286881381
// MI455X (gfx1250) — compile-verified
//
#include <hip/hip_runtime.h>
#include <cstdint>

typedef __attribute__((ext_vector_type(2))) float v2f;
typedef __attribute__((ext_vector_type(8))) float v8f;

// ---- problem constants ----
constexpr int Bn = 2;      // batch
constexpr int Dm = 96;     // d_model
constexpr int Hs = 56, Ws = 56;
constexpr int Ls = Hs * Ws;       // 3136
constexpr int Es = 192;    // d_inner
constexpr int Ns = 16;     // d_state
constexpr int Rs = 6;      // dt_rank
constexpr int PD = 38;     // dt_rank + 2*d_state
constexpr int ML = Bn * Ls;       // 6272 token rows
constexpr int CHK  = 28;   // scan chunks per sequence
constexpr int CLEN = Ls / CHK;    // 112 steps per chunk

// ws layout (floats)
constexpr size_t OFF_XBT   = 0;                             // (B,E,L)
constexpr size_t OFF_Z     = OFF_XBT   + (size_t)Bn*Es*Ls;  // (B*L,E)
constexpr size_t OFF_XCONV = OFF_Z     + (size_t)Bn*Ls*Es;  // (B,E,L)
constexpr size_t OFF_PROJ  = OFF_XCONV + (size_t)Bn*Es*Ls;  // (4,B*L,38)
constexpr size_t OFF_YS    = OFF_PROJ  + (size_t)4*ML*PD;   // (4,B,E,L)
constexpr size_t OFF_YG    = OFF_YS    + (size_t)4*Bn*Es*Ls;// (B*L,E)
constexpr size_t SCN       = (size_t)4*Bn*CHK*Es*Ns;        // per scan buffer
constexpr size_t OFF_PB    = OFF_YG    + (size_t)Bn*Ls*Es;  // chunk products P
constexpr size_t OFF_QB    = OFF_PB    + SCN;               // chunk local states q
constexpr size_t OFF_HIN   = OFF_QB    + SCN;               // chunk incoming states

static __device__ __forceinline__ v8f wmma_f32(v2f a, v2f b, v8f c) {
  // V_WMMA_F32_16X16X4_F32 : D = A(16x4) * B(4x16) + C, all fp32
  return __builtin_amdgcn_wmma_f32_16x16x4_f32(
      false, a, false, b, (short)0, c, false, false);
}

// ---------------- GEMM 1: in_proj (M=6272, K=96, N=384), 4 N-tiles/wave ----
// nblk 0..2 -> xb half (written transposed as (B,E,L)); nblk 3..5 -> z (B*L,E)
__global__ void k_inproj(const float* __restrict__ x, const float* __restrict__ w,
                         float* __restrict__ xb_t, float* __restrict__ z) {
  const int wid  = threadIdx.x >> 5;
  const int lane = threadIdx.x & 31;
  const int half = lane >> 4;
  const int lr   = lane & 15;
  const int wg   = blockIdx.x * 8 + wid;          // 392*6 = 2352 wave-tiles
  const int tileM = wg / 6, nblk = wg % 6;
  const int m0 = tileM * 16, n0 = nblk * 64;
  const int b  = m0 / Ls;
  const int l0 = m0 % Ls;                          // Ls % 16 == 0
  const float* abase = x + (size_t)b * Dm * Ls + (l0 + lr);
  v8f c[4] = {};
  for (int kk = 0; kk < Dm; kk += 4) {
    const int ka = kk + 2 * half;
    v2f a;
    a.x = abase[(size_t)ka * Ls];
    a.y = abase[(size_t)(ka + 1) * Ls];
    const float* w0 = w + (size_t)ka * 384 + n0 + lr;
    const float* w1 = w + (size_t)(ka + 1) * 384 + n0 + lr;
#pragma unroll
    for (int j = 0; j < 4; ++j) {
      v2f bb; bb.x = w0[j * 16]; bb.y = w1[j * 16];
      c[j] = wmma_f32(a, bb, c[j]);
    }
  }
  const int lbase = l0 + 8 * half;                 // C rows m = m0 + r + 8*half
  if (n0 < Es) {                                   // wave-uniform branch
#pragma unroll
    for (int j = 0; j < 4; ++j) {
      float* dst = xb_t + ((size_t)b * Es + (n0 + j * 16 + lr)) * Ls + lbase;
#pragma unroll
      for (int r = 0; r < 8; ++r) dst[r] = c[j][r];
    }
  } else {
#pragma unroll
    for (int j = 0; j < 4; ++j) {
      float* dst = z + ((size_t)b * Ls + lbase) * Es + (n0 - Es + j * 16 + lr);
#pragma unroll
      for (int r = 0; r < 8; ++r) dst[(size_t)r * Es] = c[j][r];
    }
  }
}

// ---------------- depthwise 3x3 conv (SAME) + bias + SiLU ----------------
__global__ void k_conv(const float* __restrict__ xb_t, const float* __restrict__ cw,
                       const float* __restrict__ cb, float* __restrict__ xconv) {
  const int gid = blockIdx.x * blockDim.x + threadIdx.x;
  if (gid >= Bn * Es * Ls) return;
  const int p  = gid % Ls;
  const int be = gid / Ls;
  const int e  = be % Es;
  const int h = p / Ws, w = p % Ws;
  const float* src = xb_t + (size_t)be * Ls;
  const float* wk  = cw + e * 9;
  float acc = cb[e];
#pragma unroll
  for (int di = -1; di <= 1; ++di)
#pragma unroll
    for (int dj = -1; dj <= 1; ++dj) {
      const int hh = h + di, ww = w + dj;
      if (hh >= 0 && hh < Hs && ww >= 0 && ww < Ws)
        acc += wk[(di + 1) * 3 + (dj + 1)] * src[hh * Ws + ww];
    }
  xconv[gid] = acc / (1.0f + __expf(-acc));   // SiLU
}

// ---------------- GEMM 2: x_dbl per direction (M=6272,K=192,N=38->48) -----
// one wave covers all 3 N-tiles of its M-tile; only tile j=2 is partial
__global__ void k_xdbl(const float* __restrict__ xconv, const float* __restrict__ xpw,
                       float* __restrict__ proj) {
  const int k    = blockIdx.y;
  const int wid  = threadIdx.x >> 5;
  const int lane = threadIdx.x & 31;
  const int half = lane >> 4;
  const int lr   = lane & 15;
  const int wg   = blockIdx.x * 8 + wid;          // 392 M-tiles
  const int m0 = wg * 16;
  const int b  = m0 / Ls;
  const int p0 = m0 % Ls;
  const float* abase = xconv + (size_t)b * Es * Ls + (p0 + lr);
  const float* wbase = xpw + (size_t)k * Es * PD;
  v8f c[3] = {};
  for (int kk = 0; kk < Es; kk += 4) {
    const int ka = kk + 2 * half;
    v2f a;
    a.x = abase[(size_t)ka * Ls];
    a.y = abase[(size_t)(ka + 1) * Ls];
    const float* q0 = wbase + ka * PD + lr;
    const float* q1 = wbase + (ka + 1) * PD + lr;
    v2f b0; b0.x = q0[0];  b0.y = q1[0];
    v2f b1; b1.x = q0[16]; b1.y = q1[16];
    v2f b2;                                        // cols 32..47: valid only lr<6
    b2.x = (lr < PD - 32) ? q0[32] : 0.0f;
    b2.y = (lr < PD - 32) ? q1[32] : 0.0f;
    c[0] = wmma_f32(a, b0, c[0]);
    c[1] = wmma_f32(a, b1, c[1]);
    c[2] = wmma_f32(a, b2, c[2]);
  }
  float* pbase = proj + ((size_t)k * ML + m0 + 8 * half) * PD + lr;
#pragma unroll
  for (int j = 0; j < 2; ++j)
#pragma unroll
    for (int r = 0; r < 8; ++r)
      pbase[(size_t)r * PD + j * 16] = c[j][r];
  if (lr < PD - 32) {
#pragma unroll
    for (int r = 0; r < 8; ++r)
      pbase[(size_t)r * PD + 32] = c[2][r];
  }
}

// ---------------- chunked selective scan ----------------------------------
static __device__ __forceinline__ int map_pos(int k, int t) {
  switch (k) {
    case 0: return t;
    case 1: return (Ls - 1) - t;
    case 2: return (t % Hs) * Ws + (t / Hs);
    default: { const int u = (Ls - 1) - t; return (u % Hs) * Ws + (u / Hs); }
  }
}

static __device__ __forceinline__ float softplus_dt(const float* row, const float* dtw,
                                                    float dtb) {
  float dt = dtb;
#pragma unroll
  for (int r = 0; r < Rs; ++r) dt += row[r] * dtw[r];
  return (dt > 20.0f) ? dt : log1pf(__expf(dt));
}

// pass 1: per-chunk summaries  P = prod(dA), q = local end state (h_in = 0)
__global__ void k_scan_part(const float* __restrict__ proj, const float* __restrict__ xconv,
                            const float* __restrict__ dtw_all, const float* __restrict__ dtb_all,
                            const float* __restrict__ A_log,
                            float* __restrict__ Pbuf, float* __restrict__ Qbuf) {
  const int c  = blockIdx.x % CHK;
  const int kb = blockIdx.x / CHK;
  const int b  = kb & 1, k = kb >> 1;
  const int e  = threadIdx.x;
  float An[Ns];
#pragma unroll
  for (int n = 0; n < Ns; ++n) An[n] = -__expf(A_log[e * Ns + n]);
  float dtw[Rs];
#pragma unroll
  for (int r = 0; r < Rs; ++r) dtw[r] = dtw_all[((size_t)k * Es + e) * Rs + r];
  const float dtb = dtb_all[k * Es + e];
  float h[Ns], P[Ns];
#pragma unroll
  for (int n = 0; n < Ns; ++n) { h[n] = 0.0f; P[n] = 1.0f; }
  const float* projk = proj + ((size_t)k * ML + (size_t)b * Ls) * PD;
  const float* xcb   = xconv + ((size_t)b * Es + e) * Ls;
  const int t0 = c * CLEN;
  for (int t = t0; t < t0 + CLEN; ++t) {
    const int p = map_pos(k, t);
    const float* row = projk + (size_t)p * PD;
    if (t + 1 < t0 + CLEN) __builtin_prefetch(projk + (size_t)map_pos(k, t + 1) * PD, 0, 0);
    const float dt  = softplus_dt(row, dtw, dtb);
    const float dtx = dt * xcb[p];
#pragma unroll
    for (int n = 0; n < Ns; ++n) {
      const float dA = __expf(dt * An[n]);
      P[n] *= dA;
      h[n] = dA * h[n] + dtx * row[Rs + n];
    }
  }
  const size_t base = ((((size_t)(k * Bn + b) * CHK + c) * Es + e) * Ns);
#pragma unroll
  for (int n = 0; n < Ns; ++n) { Pbuf[base + n] = P[n]; Qbuf[base + n] = h[n]; }
}

// pass 2: sequential combine over the 28 chunk summaries
__global__ void k_scan_combine(const float* __restrict__ Pbuf, const float* __restrict__ Qbuf,
                               float* __restrict__ Hin) {
  const int b = blockIdx.x & 1, k = blockIdx.x >> 1;
  const int e = threadIdx.x;
  float h[Ns];
#pragma unroll
  for (int n = 0; n < Ns; ++n) h[n] = 0.0f;
  for (int c = 0; c < CHK; ++c) {
    const size_t base = ((((size_t)(k * Bn + b) * CHK + c) * Es + e) * Ns);
#pragma unroll
    for (int n = 0; n < Ns; ++n) {
      Hin[base + n] = h[n];
      h[n] = Pbuf[base + n] * h[n] + Qbuf[base + n];
    }
  }
}

// pass 3: replay each chunk with correct incoming state; emit y at spatial p
__global__ void k_scan_final(const float* __restrict__ proj, const float* __restrict__ xconv,
                             const float* __restrict__ dtw_all, const float* __restrict__ dtb_all,
                             const float* __restrict__ A_log, const float* __restrict__ Dskip,
                             const float* __restrict__ Hin, float* __restrict__ ys) {
  const int c  = blockIdx.x % CHK;
  const int kb = blockIdx.x / CHK;
  const int b  = kb & 1, k = kb >> 1;
  const int e  = threadIdx.x;
  float An[Ns];
#pragma unroll
  for (int n = 0; n < Ns; ++n) An[n] = -__expf(A_log[e * Ns + n]);
  float dtw[Rs];
#pragma unroll
  for (int r = 0; r < Rs; ++r) dtw[r] = dtw_all[((size_t)k * Es + e) * Rs + r];
  const float dtb = dtb_all[k * Es + e];
  const float dsk = Dskip[e];
  float h[Ns];
  const size_t base = ((((size_t)(k * Bn + b) * CHK + c) * Es + e) * Ns);
#pragma unroll
  for (int n = 0; n < Ns; ++n) h[n] = Hin[base + n];
  const float* projk = proj + ((size_t)k * ML + (size_t)b * Ls) * PD;
  const float* xcb   = xconv + ((size_t)b * Es + e) * Ls;
  float* ysb         = ys + (size_t)((k * Bn + b) * Es + e) * Ls;
  const int t0 = c * CLEN;
  for (int t = t0; t < t0 + CLEN; ++t) {
    const int p = map_pos(k, t);
    const float* row = projk + (size_t)p * PD;
    if (t + 1 < t0 + CLEN) __builtin_prefetch(projk + (size_t)map_pos(k, t + 1) * PD, 0, 0);
    const float dt  = softplus_dt(row, dtw, dtb);
    const float xv  = xcb[p];
    const float dtx = dt * xv;
    float y = 0.0f;
#pragma unroll
    for (int n = 0; n < Ns; ++n) {
      const float dA = __expf(dt * An[n]);
      h[n] = dA * h[n] + dtx * row[Rs + n];
      y   += h[n] * row[Rs + Ns + n];
    }
    ysb[p] = y + dsk * xv;
  }
}

// ---------------- combine 4 dirs + LayerNorm(E) + SiLU gate ----------------
__global__ void k_norm(const float* __restrict__ ys, const float* __restrict__ z,
                       const float* __restrict__ gamma, const float* __restrict__ beta,
                       float* __restrict__ yg) {
  __shared__ float s1[Es];
  __shared__ float s2[Es];
  const int bp = blockIdx.x;
  const int e  = threadIdx.x;
  const int b  = bp / Ls, p = bp % Ls;
  float y = 0.0f;
#pragma unroll
  for (int k = 0; k < 4; ++k)
    y += ys[(size_t)((k * Bn + b) * Es + e) * Ls + p];
  s1[e] = y; s2[e] = y * y;
  __syncthreads();
  if (e < 96) { s1[e] += s1[e + 96]; s2[e] += s2[e + 96]; }
  __syncthreads();
  for (int st = 48; st > 0; st >>= 1) {
    if (e < st) { s1[e] += s1[e + st]; s2[e] += s2[e + st]; }
    __syncthreads();
  }
  const float mean = s1[0] * (1.0f / Es);
  const float var  = s2[0] * (1.0f / Es) - mean * mean;
  const float inv  = rsqrtf(var + 1e-5f);
  const float v    = (y - mean) * inv * gamma[e] + beta[e];
  const float zv   = z[(size_t)bp * Es + e];
  yg[(size_t)bp * Es + e] = v * (zv / (1.0f + __expf(-zv)));
}

// ---------------- GEMM 3: out_proj (M=6272,K=192,N=96), 6 N-tiles/wave -----
__global__ void k_outproj(const float* __restrict__ yg, const float* __restrict__ w,
                          float* __restrict__ out) {
  const int wid  = threadIdx.x >> 5;
  const int lane = threadIdx.x & 31;
  const int half = lane >> 4;
  const int lr   = lane & 15;
  const int wg   = blockIdx.x * 8 + wid;          // 392 M-tiles
  const int m0 = wg * 16;
  const int b  = m0 / Ls;                          // tile within one batch
  const int l0 = m0 % Ls;
  const float* abase = yg + (size_t)(m0 + lr) * Es;
  v8f c[6] = {};
  for (int kk = 0; kk < Es; kk += 4) {
    const int ka = kk + 2 * half;                  // even -> 8B aligned
    const v2f a = *(const v2f*)(abase + ka);
    const float* w0 = w + (size_t)ka * Dm + lr;
    const float* w1 = w + (size_t)(ka + 1) * Dm + lr;
#pragma unroll
    for (int j = 0; j < 6; ++j) {
      v2f bb; bb.x = w0[j * 16]; bb.y = w1[j * 16];
      c[j] = wmma_f32(a, bb, c[j]);
    }
  }
  const int lbase = l0 + 8 * half;
#pragma unroll
  for (int j = 0; j < 6; ++j) {
    float* dst = out + ((size_t)b * Dm + (j * 16 + lr)) * Ls + lbase;
#pragma unroll
    for (int r = 0; r < 8; ++r) dst[r] = c[j][r];
  }
}

extern "C" void kernel_launch(void* const* d_in, const int* in_sizes, int n_in,
                              void* d_out, int out_size, void* d_ws, size_t ws_size,
                              hipStream_t stream) {
  const float* x      = (const float*)d_in[0];
  const float* in_w   = (const float*)d_in[1];
  const float* conv_w = (const float*)d_in[2];
  const float* conv_b = (const float*)d_in[3];
  const float* xpw    = (const float*)d_in[4];
  const float* dtw    = (const float*)d_in[5];
  const float* dtb    = (const float*)d_in[6];
  const float* A_log  = (const float*)d_in[7];
  const float* Dskip  = (const float*)d_in[8];
  const float* ln_g   = (const float*)d_in[9];
  const float* ln_b   = (const float*)d_in[10];
  const float* out_w  = (const float*)d_in[11];
  float* out = (float*)d_out;

  float* ws    = (float*)d_ws;
  float* xb_t  = ws + OFF_XBT;
  float* zbuf  = ws + OFF_Z;
  float* xconv = ws + OFF_XCONV;
  float* proj  = ws + OFF_PROJ;
  float* ysb   = ws + OFF_YS;
  float* yg    = ws + OFF_YG;
  float* Pbuf  = ws + OFF_PB;
  float* Qbuf  = ws + OFF_QB;
  float* Hin   = ws + OFF_HIN;

  k_inproj<<<294, 256, 0, stream>>>(x, in_w, xb_t, zbuf);
  k_conv<<<(Bn * Es * Ls) / 256, 256, 0, stream>>>(xb_t, conv_w, conv_b, xconv);
  k_xdbl<<<dim3(49, 4), 256, 0, stream>>>(xconv, xpw, proj);
  k_scan_part<<<4 * Bn * CHK, Es, 0, stream>>>(proj, xconv, dtw, dtb, A_log, Pbuf, Qbuf);
  k_scan_combine<<<4 * Bn, Es, 0, stream>>>(Pbuf, Qbuf, Hin);
  k_scan_final<<<4 * Bn * CHK, Es, 0, stream>>>(proj, xconv, dtw, dtb, A_log, Dskip, Hin, ysb);
  k_norm<<<Bn * Ls, Es, 0, stream>>>(ysb, zbuf, ln_g, ln_b, yg);
  k_outproj<<<49, 256, 0, stream>>>(yg, out_w, out);
}